// SGNSLoss_56530359550797
// MI455X (gfx1250) — compile-verified
//
#include <hip/hip_runtime.h>

typedef float v2f __attribute__((ext_vector_type(2)));
typedef float v8f __attribute__((ext_vector_type(8)));

#define DIM        256
#define NSAMP      5
#define SMP_ROWS   6          // 5 sample rows + 1 zero row for padded WMMA columns
#define SMP_STRIDE 258        // padded row stride (floats); keeps b64 LDS reads 8B-aligned

__global__ __launch_bounds__(64) void sgns_zero_kernel(float* out) {
    if (threadIdx.x == 0) out[0] = 0.0f;
}

__global__ __launch_bounds__(256) void sgns_loss_kernel(
    const float* __restrict__ ctx,      // [B, 256]
    const float* __restrict__ tgt,      // [B, 256]
    const float* __restrict__ emb,      // [V, 256]
    const int*   __restrict__ neg_idx,  // [5]
    float* __restrict__ out,            // [1]
    int B)
{
    __shared__ float smp[SMP_ROWS * SMP_STRIDE];
    const int tid = threadIdx.x;

    // Zero LDS staging (row 5 stays zero: feeds WMMA B-matrix columns 5..15),
    // then gather the 5 sampled embedding rows once per block.
    for (int i = tid; i < SMP_ROWS * SMP_STRIDE; i += 256) smp[i] = 0.0f;
    __syncthreads();
    #pragma unroll
    for (int s = 0; s < NSAMP; ++s) {
        int r = neg_idx[s];
        smp[s * SMP_STRIDE + tid] = emb[(size_t)r * DIM + tid];   // blockDim == DIM
    }
    __syncthreads();

    const int lane = tid & 31;
    const int wave = tid >> 5;          // 8 waves per block
    const int col  = lane & 15;         // WMMA column / row-in-tile
    const int hi   = lane >> 4;         // half-wave select (K phase)

    const int   nTiles = B >> 4;        // 16 batch rows per wave-tile
    const int   tile0  = blockIdx.x * 8 + wave;
    const int   tStep  = gridDim.x * 8;

    // B-matrix fragment source: column `col` = sample row, cols >=5 -> zero row.
    const float* sptr = &smp[(col < NSAMP ? col : NSAMP) * SMP_STRIDE + 2 * hi];

    float lossAcc = 0.0f;

    for (int t = tile0; t < nTiles; t += tStep) {
        const int    row  = (t << 4) + col;               // this lane's batch row
        const float* cptr = ctx + (size_t)row * DIM + 2 * hi;
        const float* tptr = tgt + (size_t)row * DIM + 2 * hi;

        v8f   acc = {};     // 16x16 f32 accumulator: acc[j] = <ctx_row(j|j+8), sample_col>
        float pos = 0.0f;   // per-lane partial of the paired positive dot

        for (int k = 0; k < DIM; k += 4) {
            // A fragment: lanes 0-15 hold K=k,k+1 ; lanes 16-31 hold K=k+2,k+3 (ISA 16x4 f32 layout)
            v2f a  = __builtin_nontemporal_load((const v2f*)(cptr + k));
            v2f tv = __builtin_nontemporal_load((const v2f*)(tptr + k));
            // B fragment from LDS (mirrored layout), columns = negative samples
            v2f b  = *(const v2f*)(sptr + k);

            pos = fmaf(a.x, tv.x, pos);
            pos = fmaf(a.y, tv.y, pos);

            // D = A(16x4) * B(4x16) + C ; accumulate over K
            acc = __builtin_amdgcn_wmma_f32_16x16x4_f32(
                      false, a, false, b, (short)0, acc, false, false);
        }

        // Positive term: combine the two half-wave K-phases -> full dot in lanes 0..15.
        float dotFull = pos + __shfl_xor(pos, 16, 32);
        if (hi == 0) {
            float s = 1.0f / (1.0f + __expf(-dotFull));       // sigmoid(<c,t>)
            lossAcc += __logf(fmaxf(s, 1e-9f));
        }

        // Negative terms: every valid accumulator element is one (row, sample) dot,
        // each exactly once across lanes/VGPRs -> no cross-lane reduction needed.
        if (col < NSAMP) {
            #pragma unroll
            for (int j = 0; j < 8; ++j) {
                float x = acc[j];                              // <c_row, sample_col>
                float s = 1.0f / (1.0f + __expf(x));           // sigmoid(-x)
                lossAcc += __logf(fmaxf(s, 0.75f));
            }
        }
    }

    // Wave32 tree reduction, one atomic per wave.
    #pragma unroll
    for (int off = 16; off > 0; off >>= 1)
        lossAcc += __shfl_xor(lossAcc, off, 32);
    if (lane == 0) atomicAdd(out, lossAcc);
}

extern "C" void kernel_launch(void* const* d_in, const int* in_sizes, int n_in,
                              void* d_out, int out_size, void* d_ws, size_t ws_size,
                              hipStream_t stream) {
    const float* ctx = (const float*)d_in[0];
    const float* tgt = (const float*)d_in[1];
    const float* emb = (const float*)d_in[2];
    const int*   neg = (const int*)d_in[3];
    float*       out = (float*)d_out;

    const int B      = in_sizes[0] / DIM;     // 262144
    const int nTiles = B >> 4;                // 16384 wave-tiles
    int blocks = (nTiles + 7) / 8;            // 8 waves per block
    if (blocks > 4096) blocks = 4096;         // grid-stride covers the rest
    if (blocks < 1) blocks = 1;

    sgns_zero_kernel<<<1, 64, 0, stream>>>(out);
    sgns_loss_kernel<<<blocks, 256, 0, stream>>>(ctx, tgt, emb, neg, out, B);
}